// ControlFlowScanInplace_153705_22445499089119
// MI455X (gfx1250) — compile-verified
//
#include <hip/hip_runtime.h>

typedef __attribute__((ext_vector_type(2))) float v2f;
typedef __attribute__((ext_vector_type(8))) float v8f;

#define N_DIM 8192
#define K_DIM 256
#define TILE  128
#define LDS_STRIDE 260   // 256 + 4 floats pad: keeps 16B alignment, kills bank conflicts

extern __shared__ float smem[];

__global__ __launch_bounds__(256)
void pdist2_wmma_f32(const float* __restrict__ X,
                     const float* __restrict__ Y,
                     float* __restrict__ Z) {
    float* sx  = smem;                       // 128 x 260
    float* sy  = sx + TILE * LDS_STRIDE;     // 128 x 260
    float* sxq = sy + TILE * LDS_STRIDE;     // 128 (||x||^2)
    float* syq = sxq + TILE;                 // 128 (||y||^2)

    const int tid  = threadIdx.x;
    const int lane = tid & 31;
    const int wave = tid >> 5;

    const int rowBase = blockIdx.y * TILE;
    const int colBase = blockIdx.x * TILE;

    // ---- Stage full-K x/y tiles into LDS (float4 streams, fully coalesced) ----
    {
        const float4* gx = (const float4*)(X + (size_t)rowBase * K_DIM);
        const float4* gy = (const float4*)(Y + (size_t)colBase * K_DIM);
        for (int idx = tid; idx < TILE * (K_DIM / 4); idx += 256) {
            const int r  = idx >> 6;       // / (K_DIM/4)
            const int cv = idx & 63;
            float4 vx = gx[r * (K_DIM / 4) + cv];
            float4 vy = gy[r * (K_DIM / 4) + cv];
            *(float4*)(sx + r * LDS_STRIDE + cv * 4) = vx;
            *(float4*)(sy + r * LDS_STRIDE + cv * 4) = vy;
        }
    }
    __syncthreads();

    // ---- Squared norms for the epilogue (one row per thread) ----
    {
        const float* row = (tid < TILE) ? (sx + tid * LDS_STRIDE)
                                        : (sy + (tid - TILE) * LDS_STRIDE);
        float s = 0.f;
        #pragma unroll 8
        for (int k = 0; k < K_DIM; ++k) s += row[k] * row[k];
        if (tid < TILE) sxq[tid] = s; else syq[tid - TILE] = s;
    }
    __syncthreads();

    // ---- WMMA compute: wave grid 2(M) x 4(N); each wave owns a 64x32 patch ----
    const int wm = wave >> 2;   // 0..1  -> 64-row slab
    const int wn = wave & 3;    // 0..3  -> 32-col slab

    v8f acc[4][2];
    #pragma unroll
    for (int i = 0; i < 4; ++i)
        #pragma unroll
        for (int j = 0; j < 2; ++j) acc[i][j] = (v8f){};

    // A frag (16x4 f32): lane l -> row l%16, K = k0 + 2*(l/16) + {0,1}
    // B frag (4x16 f32): lane l -> col l%16, K = k0 + 2*(l/16) + {0,1} (y is row-major in N)
    const int mrow  = lane & 15;
    const int khalf = (lane >> 4) * 2;
    const float* aBase = sx + (wm * 64 + mrow) * LDS_STRIDE + khalf;
    const float* bBase = sy + (wn * 32 + mrow) * LDS_STRIDE + khalf;

    #pragma unroll 4
    for (int k0 = 0; k0 < K_DIM; k0 += 4) {
        v2f a[4], b[2];
        #pragma unroll
        for (int i = 0; i < 4; ++i)
            a[i] = *(const v2f*)(aBase + i * 16 * LDS_STRIDE + k0);
        #pragma unroll
        for (int j = 0; j < 2; ++j)
            b[j] = *(const v2f*)(bBase + j * 16 * LDS_STRIDE + k0);
        #pragma unroll
        for (int i = 0; i < 4; ++i)
            #pragma unroll
            for (int j = 0; j < 2; ++j)
                acc[i][j] = __builtin_amdgcn_wmma_f32_16x16x4_f32(
                    /*neg_a=*/false, a[i], /*neg_b=*/false, b[j],
                    /*c_mod=*/(short)0, acc[i][j],
                    /*reuse_a=*/false, /*reuse_b=*/false);
    }

    // ---- Epilogue: z = ||x||^2 + ||y||^2 - 2*cross, streamed to HBM ----
    const int hi = lane >> 4;
    #pragma unroll
    for (int i = 0; i < 4; ++i) {
        #pragma unroll
        for (int j = 0; j < 2; ++j) {
            const int lcol = wn * 32 + j * 16 + (lane & 15);
            const float yq = syq[lcol];
            #pragma unroll
            for (int r = 0; r < 8; ++r) {
                const int lrow = wm * 64 + i * 16 + r + 8 * hi;
                const float v = sxq[lrow] + yq - 2.0f * acc[i][j][r];
                Z[(size_t)(rowBase + lrow) * N_DIM + (colBase + lcol)] = v;
            }
        }
    }
}

extern "C" void kernel_launch(void* const* d_in, const int* in_sizes, int n_in,
                              void* d_out, int out_size, void* d_ws, size_t ws_size,
                              hipStream_t stream) {
    (void)in_sizes; (void)n_in; (void)d_ws; (void)ws_size; (void)out_size;
    const float* x = (const float*)d_in[0];
    const float* y = (const float*)d_in[1];
    float* z = (float*)d_out;

    const size_t smemBytes = (size_t)(2 * TILE * LDS_STRIDE + 2 * TILE) * sizeof(float);
    static_assert(2 * TILE * LDS_STRIDE * sizeof(float) + 2 * TILE * sizeof(float) <= 320 * 1024,
                  "fits WGP LDS");

    // Opt in to >64KB dynamic LDS (host-side, not a stream op; capture-safe).
    hipFuncSetAttribute((const void*)pdist2_wmma_f32,
                        hipFuncAttributeMaxDynamicSharedMemorySize, (int)smemBytes);

    dim3 grid(N_DIM / TILE, N_DIM / TILE);  // 64 x 64
    dim3 block(256);
    pdist2_wmma_f32<<<grid, block, smemBytes, stream>>>(x, y, z);
}